// VectorQuantizer_74423193305695
// MI455X (gfx1250) — compile-verified
//
#include <hip/hip_runtime.h>
#include <math.h>

// CDNA5 / gfx1250 vector-quantizer.
// Core op: V_WMMA_F32_16X16X4_F32  (D = A(16x4,f32) * B(4x16,f32) + C(16x16,f32))
// Two independent accumulator chains per chunk to break the C->C RAW chain
// on the matrix pipe.

typedef __attribute__((ext_vector_type(2))) float v2f;
typedef __attribute__((ext_vector_type(8))) float v8f;

#define VQ_B 131072
#define VQ_D 64
#define VQ_K 4096

__global__ __launch_bounds__(256) void VectorQuantizer_74423193305695_kernel(
    const float* __restrict__ z_e,   // (B, 64)
    const float* __restrict__ cb,    // (K, 64)
    float* __restrict__ z_q,         // (B, 64)
    int* __restrict__ codes)         // (B,)
{
    const int lane = threadIdx.x & 31;
    const int wave = threadIdx.x >> 5;
    const int l16  = lane & 15;
    const int half = lane >> 4;          // 0: lanes 0-15, 1: lanes 16-31
    const int row_base = (blockIdx.x * 8 + wave) * 16;

    // ---- Load A tile (16 rows x 64 cols of z_e), resident for whole K loop.
    // WMMA f32 16x4 A-layout: lane<16 -> (M=lane, K=4i+{0,1}); lane>=16 -> K=4i+{2,3}
    v2f a[16];
    {
        const float* zrow = z_e + (size_t)(row_base + l16) * VQ_D + 2 * half;
#pragma unroll
        for (int i = 0; i < 16; ++i)
            a[i] = *(const v2f*)(zrow + 4 * i);
    }

    // Per-lane running argmin. Lane's column is (chunk_base + l16); accumulator
    // element j corresponds to row (j + 8*half) of the 16-row tile.
    float best[8];
    int   bidx[8];
#pragma unroll
    for (int j = 0; j < 8; ++j) { best[j] = INFINITY; bidx[j] = 0; }

    for (int col = 0; col < VQ_K; col += 16) {
        const float* crow = cb + (size_t)(col + l16) * VQ_D + 2 * half;

        // Prefetch next chunk's rows (speculative; dropped if OOB).
        __builtin_prefetch(crow + 16 * VQ_D, 0, 0);

        // ---- Load B tile (codebook^T, 64x16). Mirrored layout: this lane
        // holds codebook row (col + l16), k-pair {4i+2*half, 4i+2*half+1}.
        v2f b[16];
#pragma unroll
        for (int i = 0; i < 16; ++i)
            b[i] = *(const v2f*)(crow + 4 * i);

        // ---- ||e||^2 partials for this lane's column (overlaps WMMA chain).
        float e2p0 = 0.f, e2p1 = 0.f;
#pragma unroll
        for (int i = 0; i < 16; i += 2) {
            e2p0 = __builtin_fmaf(b[i].x, b[i].x,
                   __builtin_fmaf(b[i].y, b[i].y, e2p0));
            e2p1 = __builtin_fmaf(b[i + 1].x, b[i + 1].x,
                   __builtin_fmaf(b[i + 1].y, b[i + 1].y, e2p1));
        }
        const float e2p = e2p0 + e2p1;
        const float e2 = e2p + __shfl_xor(e2p, 16, 32);

        // ---- 16 f32 WMMAs over D=64, split into two independent chains.
        v8f c0 = {};
        v8f c1 = {};
#pragma unroll
        for (int i = 0; i < 16; i += 2) {
            c0 = __builtin_amdgcn_wmma_f32_16x16x4_f32(
                     false, a[i],     false, b[i],     (short)0, c0, false, false);
            c1 = __builtin_amdgcn_wmma_f32_16x16x4_f32(
                     false, a[i + 1], false, b[i + 1], (short)0, c1, false, false);
        }

        // ---- score = ||e||^2 - 2*(z.e)   (||z||^2 dropped: constant per row)
        const int myidx = col + l16;
#pragma unroll
        for (int j = 0; j < 8; ++j) {
            const float zc = c0[j] + c1[j];
            const float s = __builtin_fmaf(-2.f, zc, e2);
            if (s < best[j]) { best[j] = s; bidx[j] = myidx; }
        }
    }

    // ---- Cross-lane argmin over the 16 lanes of each half (xor masks stay
    // within the 16-lane group). Tie -> smaller index (jnp.argmin semantics).
#pragma unroll
    for (int m = 8; m >= 1; m >>= 1) {
#pragma unroll
        for (int j = 0; j < 8; ++j) {
            const float ov = __shfl_xor(best[j], m, 32);
            const int   oi = __shfl_xor(bidx[j], m, 32);
            if (ov < best[j] || (ov == best[j] && oi < bidx[j])) {
                best[j] = ov; bidx[j] = oi;
            }
        }
    }

    // ---- Write codes: lane 0 covers rows 0..7, lane 16 covers rows 8..15.
    if (l16 == 0) {
#pragma unroll
        for (int j = 0; j < 8; ++j)
            codes[row_base + half * 8 + j] = bidx[j];
    }

    // ---- Gather z_q = codebook[codes]: 16 rows, 2 floats per lane per row.
#pragma unroll
    for (int r = 0; r < 16; ++r) {
        const int j = r & 7;
        const int g = r >> 3;                       // source 16-lane group
        const int code = __shfl(bidx[j], g * 16, 32);
        const v2f v = *(const v2f*)(cb + (size_t)code * VQ_D + lane * 2);
        *(v2f*)(z_q + (size_t)(row_base + r) * VQ_D + lane * 2) = v;
    }
}

extern "C" void kernel_launch(void* const* d_in, const int* in_sizes, int n_in,
                              void* d_out, int out_size, void* d_ws, size_t ws_size,
                              hipStream_t stream) {
    (void)in_sizes; (void)n_in; (void)d_ws; (void)ws_size; (void)out_size;

    const float* z_e = (const float*)d_in[0];           // (B, 64) f32
    const float* cb  = (const float*)d_in[1];           // (K, 64) f32

    float* z_q  = (float*)d_out;                                    // (B, 64) f32
    int*   codes = (int*)((float*)d_out + (size_t)VQ_B * VQ_D);     // (B,) int32

    // 8 waves/block * 16 rows/wave = 128 rows per block
    const int blocks = VQ_B / 128;   // 1024
    VectorQuantizer_74423193305695_kernel<<<blocks, 256, 0, stream>>>(
        z_e, cb, z_q, codes);
}